// Net_38345468019358
// MI455X (gfx1250) — compile-verified
//
#include <hip/hip_runtime.h>

// TV-ROF Chambolle-Pock primal-dual, 10 iterations, 4096x4096 fp32.
// Bandwidth-bound stencil: all global traffic b128-vectorized (4 px/thread),
// x_tilde/img tiles staged through LDS via CDNA5 async VMEM->LDS copies
// (global_load_async_to_lds_b128, ASYNCcnt, s_wait_asynccnt).

#define IMH 4096
#define IMW 4096
#define NPIX (IMH * IMW)

#define SIGMA_C  14.285714285714286f   // 1/(7*0.01)
#define TAU_C    0.01f
#define THETA_C  0.5f
#define LT_C     0.04f                 // LAMBDA_ROF * TAU
#define INV_DEN  0.9615384615384615f   // 1/(1+LAMBDA_ROF*TAU)

// ---- dual-kernel tile geometry (4 px per thread) ----
#define BX 32
#define BY 8
#define VW 4
#define TCOLS (BX * VW + VW)        // 132 floats (33 x 16B chunks, 16B-aligned rows)
#define TROWS (BY + 1)              // 9
#define TCHNK (TCOLS / 4)           // 33 b128 chunks per row
#define TLN   (TROWS * TCOLS)       // 1188 floats per tile

__device__ __forceinline__ float4 ld4(const float* p) {
    return *reinterpret_cast<const float4*>(p);
}
__device__ __forceinline__ void st4(float* p, float4 v) {
    *reinterpret_cast<float4*>(p) = v;
}
__device__ __forceinline__ float clip1(float v) {
    return fminf(fmaxf(v, -1.0f), 1.0f);
}

__device__ __forceinline__ void async_tile_load_b128(
    float* lds_tile, const float* __restrict__ src,
    int i0, int j0, int tid)
{
    // Stage a TROWS x TCOLS tile (row-clamped halo) with async 16B copies.
    // Rightmost chunk of border blocks re-reads the row's last aligned 16B
    // (never reads OOB); the resulting column-4095 garbage is masked at the
    // y-store, matching the reference's structural zeroing of y0[:, -1].
    for (int k = tid; k < TROWS * TCHNK; k += BX * BY) {
        int r  = k / TCHNK;
        int ch = k - r * TCHNK;
        int gi = min(i0 + r, IMH - 1);
        int cf = j0 + ch * 4;
        if (cf > IMW - 4) cf = IMW - 4;
        unsigned voff = (unsigned)((gi * IMW + cf) * 4);
        unsigned lo   = (unsigned)(size_t)(&lds_tile[r * TCOLS + ch * 4]);
        asm volatile("global_load_async_to_lds_b128 %0, %1, %2 offset:0"
                     :: "v"(lo), "v"(voff), "s"(src) : "memory");
    }
}

// ---------------- init: y = grad(img), x = img ----------------
__global__ __launch_bounds__(256) void tv_init(
    const float* __restrict__ img,
    float* __restrict__ y0, float* __restrict__ y1,
    float* __restrict__ x)
{
    const int jv  = (blockIdx.x * 256 + threadIdx.x) * VW;
    const int i   = blockIdx.y;
    const int idx = i * IMW + jv;

    float4 c4 = ld4(img + idx);
    float  c[5] = {c4.x, c4.y, c4.z, c4.w,
                   (jv + VW < IMW) ? img[idx + VW] : c4.w};   // clamp -> gx=0 at last col
    float4 d4 = (i < IMH - 1) ? ld4(img + idx + IMW) : c4;    // clamp -> gy=0 at last row
    float  d[4] = {d4.x, d4.y, d4.z, d4.w};

    float g0[4], g1[4];
#pragma unroll
    for (int e = 0; e < 4; ++e) {
        g0[e] = c[e + 1] - c[e];
        g1[e] = d[e] - c[e];
    }
    st4(y0 + idx, make_float4(g0[0], g0[1], g0[2], g0[3]));
    st4(y1 + idx, make_float4(g1[0], g1[1], g1[2], g1[3]));
    st4(x  + idx, c4);
}

// -------- dual update: y = clip(y + SIGMA*w*grad(x_tilde), -1, 1) --------
__global__ __launch_bounds__(BX * BY) void tv_dual(
    const float* __restrict__ xt,
    const float* __restrict__ img,
    const float* __restrict__ w1p, const float* __restrict__ w2p,
    float* __restrict__ y0, float* __restrict__ y1)
{
    __shared__ float xtile[TLN];
    __shared__ float itile[TLN];

    const int tx = threadIdx.x, ty = threadIdx.y;
    const int tid = ty * BX + tx;
    const int i0 = blockIdx.y * BY;
    const int j0 = blockIdx.x * (BX * VW);

    async_tile_load_b128(xtile, xt,  i0, j0, tid);
    async_tile_load_b128(itile, img, i0, j0, tid);
    asm volatile("s_wait_asynccnt 0x0" ::: "memory");
    __syncthreads();

    const int i   = i0 + ty;
    const int tc  = tx * VW;              // tile column of first pixel
    const int idx = i * IMW + j0 + tc;

    const float a = w1p[0], b = w2p[0];

    float4 t;
    t = ld4(&xtile[ty * TCOLS + tc]);
    float xc[5] = {t.x, t.y, t.z, t.w, xtile[ty * TCOLS + tc + 4]};
    t = ld4(&xtile[(ty + 1) * TCOLS + tc]);
    float xd[4] = {t.x, t.y, t.z, t.w};

    t = ld4(&itile[ty * TCOLS + tc]);
    float ic[5] = {t.x, t.y, t.z, t.w, itile[ty * TCOLS + tc + 4]};
    t = ld4(&itile[(ty + 1) * TCOLS + tc]);
    float id[4] = {t.x, t.y, t.z, t.w};

    float4 y0v = ld4(y0 + idx);
    float4 y1v = ld4(y1 + idx);
    float y0a[4] = {y0v.x, y0v.y, y0v.z, y0v.w};
    float y1a[4] = {y1v.x, y1v.y, y1v.z, y1v.w};

    const bool right_edge = (j0 + tc + 3 == IMW - 1);
    const bool bottom     = (i == IMH - 1);

    float r0[4], r1[4];
#pragma unroll
    for (int e = 0; e < 4; ++e) {
        float gx  = xc[e + 1] - xc[e];
        float gy  = xd[e] - xc[e];                       // exact 0 at last row (row clamp)
        float gxi = ic[e + 1] - ic[e];
        float gyi = id[e] - ic[e];
        float w0v = fmaf(b, __expf(-fabsf(gxi)), a);     // w = w1 + w2*exp(-|grad(img)|)
        float w1v = fmaf(b, __expf(-fabsf(gyi)), a);
        r0[e] = clip1(fmaf(SIGMA_C * w0v, gx, y0a[e]));
        r1[e] = clip1(fmaf(SIGMA_C * w1v, gy, y1a[e]));
        if (bottom) r1[e] = 0.0f;                        // y1[-1,:] structurally zero
    }
    if (right_edge) r0[3] = 0.0f;                        // y0[:,-1] structurally zero

    st4(y0 + idx, make_float4(r0[0], r0[1], r0[2], r0[3]));
    st4(y1 + idx, make_float4(r1[0], r1[1], r1[2], r1[3]));
}

// -------- primal: x = (x + TAU*div(w*y) + LT*img)/(1+LT); xt over-relax --------
__global__ __launch_bounds__(256) void tv_primal(
    const float* __restrict__ img,
    const float* __restrict__ w1p, const float* __restrict__ w2p,
    const float* __restrict__ y0, const float* __restrict__ y1,
    float* __restrict__ x, float* __restrict__ xt)
{
    const int jv  = (blockIdx.x * 256 + threadIdx.x) * VW;
    const int i   = blockIdx.y;
    const int idx = i * IMW + jv;

    const float a = w1p[0], b = w2p[0];

    float4 c4 = ld4(img + idx);
    float  c[5] = {c4.x, c4.y, c4.z, c4.w,
                   (jv + VW < IMW) ? img[idx + VW] : c4.w};
    float4 d4 = (i < IMH - 1) ? ld4(img + idx + IMW) : c4;
    float  d[4] = {d4.x, d4.y, d4.z, d4.w};
    float4 u4 = (i > 0) ? ld4(img + idx - IMW) : c4;      // used only when i>0
    float  u[4] = {u4.x, u4.y, u4.z, u4.w};
    float  il = (jv > 0) ? img[idx - 1] : 0.0f;           // used only when jv>0

    float4 y0v = ld4(y0 + idx);
    float  y0a[5] = {(jv > 0) ? y0[idx - 1] : 0.0f, y0v.x, y0v.y, y0v.z, y0v.w};
    float4 y1v = ld4(y1 + idx);
    float  y1a[4] = {y1v.x, y1v.y, y1v.z, y1v.w};
    float4 y1u4 = (i > 0) ? ld4(y1 + idx - IMW) : make_float4(0.f, 0.f, 0.f, 0.f);
    float  y1u[4] = {y1u4.x, y1u4.y, y1u4.z, y1u4.w};

    float4 xo4 = ld4(x + idx);
    float  xo[4] = {xo4.x, xo4.y, xo4.z, xo4.w};

    float xn[4], xr[4];
#pragma unroll
    for (int e = 0; e < 4; ++e) {
        float icn = c[e];
        float w0c = fmaf(b, __expf(-fabsf(c[e + 1] - icn)), a);   // w0 at (i,j)
        float w1c = fmaf(b, __expf(-fabsf(d[e] - icn)), a);       // w1 at (i,j)
        float p = w0c * y0a[e + 1];        // y0 already 0 at j==W-1
        float q = w1c * y1a[e];            // y1 already 0 at i==H-1

        float ilv = (e == 0) ? il : c[e - 1];
        float pm = 0.0f, qm = 0.0f;
        if (jv + e > 0) {
            float w0l = fmaf(b, __expf(-fabsf(icn - ilv)), a);    // w0 at (i,j-1)
            pm = w0l * y0a[e];
        }
        if (i > 0) {
            float w1u = fmaf(b, __expf(-fabsf(icn - u[e])), a);   // w1 at (i-1,j)
            qm = w1u * y1u[e];
        }

        float dv = (p - pm) + (q - qm);
        xn[e] = fmaf(TAU_C, dv, fmaf(LT_C, icn, xo[e])) * INV_DEN;
        xr[e] = fmaf(THETA_C, xn[e] - xo[e], xn[e]);
    }
    st4(x  + idx, make_float4(xn[0], xn[1], xn[2], xn[3]));
    st4(xt + idx, make_float4(xr[0], xr[1], xr[2], xr[3]));
}

extern "C" void kernel_launch(void* const* d_in, const int* in_sizes, int n_in,
                              void* d_out, int out_size, void* d_ws, size_t ws_size,
                              hipStream_t stream) {
    // setup_inputs order: x (unused), img_obs, w1, w2
    const float* img = (const float*)d_in[1];
    const float* w1s = (const float*)d_in[2];
    const float* w2s = (const float*)d_in[3];

    float* x  = (float*)d_out;                 // x lives in d_out (in-place updates)
    float* ws = (float*)d_ws;                  // 3 planes: y0, y1, x_tilde (201 MB)
    float* y0 = ws;
    float* y1 = ws + (size_t)NPIX;
    float* xt = ws + 2 * (size_t)NPIX;

    dim3 bRow(256, 1, 1), gRow(IMW / (256 * VW), IMH, 1);   // (4, 4096)
    dim3 bA(BX, BY, 1),   gA(IMW / (BX * VW), IMH / BY, 1); // (32, 512)

    tv_init<<<gRow, bRow, 0, stream>>>(img, y0, y1, x);

    for (int it = 0; it < 10; ++it) {
        const float* xtin = (it == 0) ? img : xt;   // x_tilde starts as img_obs
        tv_dual  <<<gA,   bA,   0, stream>>>(xtin, img, w1s, w2s, y0, y1);
        tv_primal<<<gRow, bRow, 0, stream>>>(img, w1s, w2s, y0, y1, x, xt);
    }
}